// G2_update_84937273245884
// MI455X (gfx1250) — compile-verified
//
#include <hip/hip_runtime.h>

// Problem constants (from reference)
#define N_G2   1024
#define DVEC   64
#define CHUNK  32                 // i-rows per LDS tile
#define NCHUNK (N_G2 / CHUNK)     // 32 chunks
#define BLOCK  256                // 8 waves (wave32)

// gfx12+ CPol temporal hints (load): 0 = RT (regular), 1 = NT (non-temporal)
#define CPOL_RT 0
#define CPOL_NT 1

// ---------------------------------------------------------------------------
// CDNA5 async global->LDS path (gfx1250): GLOBAL_LOAD_ASYNC_TO_LDS_B128,
// tracked by ASYNCcnt, waited with s_wait_asynccnt. Guarded so the kernel
// still compiles (synchronous LDS staging) if the builtins are absent.
// ---------------------------------------------------------------------------
#if defined(__gfx1250__) &&                                                   \
    __has_builtin(__builtin_amdgcn_global_load_async_to_lds_b128) &&          \
    __has_builtin(__builtin_amdgcn_s_wait_asynccnt)
#define USE_ASYNC 1
#else
#define USE_ASYNC 0
#endif

// Builtin parameter types (from hipcc diagnostic): pointer to 4xi32 vector in
// AS1 (global, prints as __device__) and AS3 (LDS, prints as __shared__).
typedef int v4i __attribute__((__vector_size__(16)));
typedef __attribute__((address_space(1))) v4i as1_v4i;
typedef __attribute__((address_space(3))) v4i as3_v4i;

// 16B async copy. OFF is the ISA's INST_OFFSET, which is applied to BOTH the
// global address and the LDS address (ISA 08 §4.4) — so one base pointer pair
// plus an immediate covers a whole tile with zero extra address VALU.
template <int OFF, int CPOL>
__device__ __forceinline__ void cp16(const float* g, float* l) {
#if USE_ASYNC
  __builtin_amdgcn_global_load_async_to_lds_b128(
      (as1_v4i*)(uintptr_t)g,
      (as3_v4i*)(uint32_t)(uintptr_t)l,
      OFF, CPOL);
#else
  *(float4*)((char*)l + OFF) = *(const float4*)((const char*)g + OFF);
#endif
}

template <int N>
__device__ __forceinline__ void wait_async() {
#if USE_ASYNC
  __builtin_amdgcn_s_wait_asynccnt(N);
#endif
}

__global__ void __launch_bounds__(BLOCK)
g2_update_kernel(const float* __restrict__ emb,   // (n_rows, 64)
                 const float* __restrict__ W,     // (1024, 1024, 64)
                 const float* __restrict__ R,     // (1024, 1024, 64)
                 float* __restrict__ out,         // (n_rows, 64)
                 int n_rows) {
  const int j = blockIdx.x;
  const int t = threadIdx.x;

  if (j >= N_G2) {
    // Pass-through rows 1024..n_rows-1 (whole block takes this path; no
    // barriers are skipped by a subset of threads).
    if (j < n_rows && t < DVEC) {
      out[(size_t)j * DVEC + t] = emb[(size_t)j * DVEC + t];
    }
    return;
  }

  // Double-buffered streaming tiles: 3 * 2 * 32*64*4B = 48 KB (+1 KB reduce)
  __shared__ __align__(16) float Wb[2][CHUNK * DVEC];
  __shared__ __align__(16) float Rb[2][CHUNK * DVEC];
  __shared__ __align__(16) float Eb[2][CHUNK * DVEC];
  __shared__ float red[BLOCK];

  // Issue one chunk's async loads into buffer b.
  // 3 arrays * 2048 floats = 1536 b128 transfers = 6 per thread
  // => exactly 6 ASYNCcnt increments per wave per chunk.
  // W/R are streamed once -> NT hint (don't evict L2-resident emb);
  // emb is reused by every block -> RT.
  auto issue = [&](int c, int b) {
    const size_t base = ((size_t)j * N_G2 + (size_t)c * CHUNK) * DVEC;
    const float* gW = W + base;
    const float* gR = R + base;
    const float* gE = emb + (size_t)c * CHUNK * DVEC;
    const int v = t * 4;  // float index, 16B aligned; +4096B imm = 2nd half
    cp16<0,    CPOL_NT>(gW + v, &Wb[b][v]);
    cp16<4096, CPOL_NT>(gW + v, &Wb[b][v]);
    cp16<0,    CPOL_NT>(gR + v, &Rb[b][v]);
    cp16<4096, CPOL_NT>(gR + v, &Rb[b][v]);
    cp16<0,    CPOL_RT>(gE + v, &Eb[b][v]);
    cp16<4096, CPOL_RT>(gE + v, &Eb[b][v]);
  };

  const int d    = t & (DVEC - 1);  // owned feature dim
  const int isub = t >> 6;          // 0..3: i-row stripe within a chunk

  float accW = 0.0f;  // sum_i W[j,i,d] * emb[i,d]
  float accR = 0.0f;  // sum_i R[j,i,d]

  issue(0, 0);

  for (int c = 0; c < NCHUNK; ++c) {
    const int b = c & 1;
    if (c + 1 < NCHUNK) {
      issue(c + 1, b ^ 1);  // prefetch next tile into the other buffer
      wait_async<6>();      // in-order completion => chunk c's 6 ops done
    } else {
      wait_async<0>();      // drain last chunk
    }
    __syncthreads();        // chunk c visible to all waves

#pragma unroll
    for (int r = isub; r < CHUNK; r += 4) {
      const int idx = r * DVEC + d;
      accW = fmaf(Wb[b][idx], Eb[b][idx], accW);
      accR += Rb[b][idx];
    }
    __syncthreads();        // all waves done with buffer b before it is
                            // overwritten by issue(c+2) next iteration
  }

  // 4-way partial reduction over the i-stripes, then add emb[j,d].
  red[t] = accW + accR;
  __syncthreads();
  if (t < DVEC) {
    const float total = red[t] + red[t + 64] + red[t + 128] + red[t + 192];
    out[(size_t)j * DVEC + t] = total + emb[(size_t)j * DVEC + t];
  }
}

extern "C" void kernel_launch(void* const* d_in, const int* in_sizes, int n_in,
                              void* d_out, int out_size, void* d_ws, size_t ws_size,
                              hipStream_t stream) {
  (void)n_in; (void)d_ws; (void)ws_size; (void)out_size;
  const float* emb = (const float*)d_in[0];  // (2048, 64) f32
  const float* W   = (const float*)d_in[1];  // (1024, 1024, 64) f32
  const float* R   = (const float*)d_in[2];  // (1024, 1024, 64) f32
  float* out       = (float*)d_out;          // (2048, 64) f32

  const int n_rows = in_sizes[0] / DVEC;     // 2048
  g2_update_kernel<<<dim3(n_rows), dim3(BLOCK), 0, stream>>>(emb, W, R, out, n_rows);
}